// ss_Module_36644660969987
// MI455X (gfx1250) — compile-verified
//
#include <hip/hip_runtime.h>
#include <math.h>

// Problem constants (from reference): N=2, C=128, H=W=64, S=5, CK=C/8=16
static constexpr int kN   = 2;
static constexpr int kC   = 128;
static constexpr int kCK  = 16;
static constexpr int kHW  = 4096;
static constexpr int kS   = 5;
static constexpr float kEPS = 1e-5f;

typedef float v2f __attribute__((ext_vector_type(2)));
typedef float v8f __attribute__((ext_vector_type(8)));

// D(16x16 f32) = A(16x4 f32) * B(4x16 f32) + C  — CDNA5 fp32 WMMA
static __device__ __forceinline__ v8f wmma_k4(v2f a, v2f b, v8f c) {
  return __builtin_amdgcn_wmma_f32_16x16x4_f32(false, a, false, b, (short)0, c,
                                               false, false);
}

// Lane<->lane+16 swap via DS_SWIZZLE (SWAPX16: and=0x1f, xor=0x10 -> 0x401F)
static __device__ __forceinline__ float swap16(float x) {
  return __int_as_float(__builtin_amdgcn_ds_swizzle(__float_as_int(x), 0x401F));
}

// gfx1250 async global->LDS copy (ASYNCcnt-tracked), 16 bytes per lane
static __device__ __forceinline__ void async_b128(unsigned lds_off,
                                                  const float* gptr) {
  asm volatile("global_load_async_to_lds_b128 %0, %1, off"
               :: "v"(lds_off), "v"(gptr) : "memory");
}
static __device__ __forceinline__ void wait_async0() {
  asm volatile("s_wait_asynccnt 0" ::: "memory");
}

// ---------------------------------------------------------------------------
// K1: per-position softmax over stack dim s  -> att1[n][p][s]
// ---------------------------------------------------------------------------
__global__ __launch_bounds__(256) void att1_kernel(
    const float* __restrict__ query, const float* __restrict__ key_stack,
    float* __restrict__ att1) {
  int idx = blockIdx.x * 256 + threadIdx.x;  // over N*HW = 8192
  if (idx >= kN * kHW) return;
  int n = idx >> 12, p = idx & (kHW - 1);
  float e[kS] = {0.f, 0.f, 0.f, 0.f, 0.f};
  #pragma unroll
  for (int k = 0; k < kCK; ++k) {
    float qv = query[(n * kCK + k) * kHW + p];
    const float* ks = key_stack + ((size_t)(n * kCK + k) * kHW + p) * kS;
    #pragma unroll
    for (int s = 0; s < kS; ++s) e[s] += ks[s] * qv;
  }
  float mx = e[0];
  #pragma unroll
  for (int s = 1; s < kS; ++s) mx = fmaxf(mx, e[s]);
  float sum = 0.f;
  #pragma unroll
  for (int s = 0; s < kS; ++s) { e[s] = __expf(e[s] - mx); sum += e[s]; }
  float inv = 1.f / sum;
  #pragma unroll
  for (int s = 0; s < kS; ++s) att1[(size_t)idx * kS + s] = e[s] * inv;
}

// ---------------------------------------------------------------------------
// K2: out2[n][c][p] = sum_s fea_stack[n][c][p][s] * att1[n][p][s]
// ---------------------------------------------------------------------------
__global__ __launch_bounds__(256) void out2_kernel(
    const float* __restrict__ fea_stack, const float* __restrict__ att1,
    float* __restrict__ out2) {
  int idx = blockIdx.x * 256 + threadIdx.x;  // over N*C*HW = 1048576
  int p = idx & (kHW - 1);
  int c = (idx >> 12) & (kC - 1);
  int n = idx >> 19;
  const float* fs = fea_stack + ((size_t)(n * kC + c) * kHW + p) * kS;
  const float* a  = att1 + ((size_t)n * kHW + p) * kS;
  float acc = 0.f;
  #pragma unroll
  for (int s = 0; s < kS; ++s) acc += fs[s] * a[s];
  out2[idx] = acc;  // idx == (n*C + c)*HW + p
}

// ---------------------------------------------------------------------------
// K3: key2[n][k][p] = key_w[k][:] . out2[n][:][p] + key_b[k]   (tiny GEMV)
// ---------------------------------------------------------------------------
__global__ __launch_bounds__(256) void key2_kernel(
    const float* __restrict__ key_w, const float* __restrict__ key_b,
    const float* __restrict__ out2, float* __restrict__ key2) {
  int idx = blockIdx.x * 256 + threadIdx.x;  // over N*CK*HW = 131072
  int p = idx & (kHW - 1);
  int k = (idx >> 12) & (kCK - 1);
  int n = idx >> 16;
  float acc = key_b[k];
  #pragma unroll 8
  for (int c = 0; c < kC; ++c)
    acc += key_w[k * kC + c] * out2[(size_t)(n * kC + c) * kHW + p];
  key2[idx] = acc;  // idx == (n*CK + k)*HW + p
}

// ---------------------------------------------------------------------------
// K4: flash attention over keys p (softmax dim) for 128-query blocks.
//   energy[p][q] = sum_k key2[k][p] * q1[k][q];  softmax over p;
//   O[c][q] = sum_p P[p][q] * out2[c][p].
// Key dim partitioned across blockIdx.y; partials (unnormalized O, m, l) -> ws.
// Per block: 8 waves; wave w owns q-subtile w (16 cols) x all 128 c rows.
// V tile is double-buffered in LDS via async global->LDS B128 loads.
// ---------------------------------------------------------------------------
__global__ __launch_bounds__(256) void attn2_kernel(
    const float* __restrict__ q1, const float* __restrict__ key2,
    const float* __restrict__ out2, float* __restrict__ Opart,
    float* __restrict__ m_arr, float* __restrict__ l_arr, int kTiles) {
  constexpr int VST  = 20;          // LDS row stride (floats): vt[c][p], padded
  constexpr int BUFF = kC * VST;    // floats per buffer (2560)
  __shared__ float vt[2 * BUFF];    // double buffer, 20 KB

  const int n    = blockIdx.z;
  const int part = blockIdx.y;
  const int qb   = blockIdx.x;
  const int tid  = threadIdx.x;
  const int wave = tid >> 5;
  const int lane = tid & 31;
  const int l16  = lane & 15;
  const bool hi  = lane >= 16;

  const int q = qb * 128 + wave * 16 + l16;  // this lane's query column

  // Async-staging assignments: thread -> (c row, 4-float p group), x2 rows
  const int stp = (tid & 3) * 4;    // p offset within tile (0,4,8,12)
  const int stc = tid >> 2;         // c row 0..63 (second transfer: +64)
  const unsigned vbase = (unsigned)(uintptr_t)vt;
  const float* g_src = out2 + (size_t)n * kC * kHW;

  // Preload Q as B-operand chunks: vgpr pair covers K rows {k0,k0+1|k0+2,k0+3}
  v2f Bq[4];
  #pragma unroll
  for (int c4 = 0; c4 < 4; ++c4) {
    int k0 = c4 * 4 + (hi ? 2 : 0);
    Bq[c4].x = q1[(size_t)(n * kCK + k0) * kHW + q];
    Bq[c4].y = q1[(size_t)(n * kCK + k0 + 1) * kHW + q];
  }

  v8f O[8];
  #pragma unroll
  for (int t = 0; t < 8; ++t)
    #pragma unroll
    for (int j = 0; j < 8; ++j) O[t][j] = 0.f;
  float m = -INFINITY, l = 0.f;

  // Kick off tile 0 into buffer 0
  {
    const int pb0 = part * kTiles * 16;
    async_b128(vbase + (unsigned)(stc * VST + stp) * 4,
               g_src + (size_t)stc * kHW + pb0 + stp);
    async_b128(vbase + (unsigned)((stc + 64) * VST + stp) * 4,
               g_src + (size_t)(stc + 64) * kHW + pb0 + stp);
  }

  for (int t = 0; t < kTiles; ++t) {
    const int cur = t & 1;
    const int pb  = (part * kTiles + t) * 16;

    wait_async0();      // this wave's contributions to tile t are in LDS
    __syncthreads();    // everyone's tile-t data visible; prev reads retired

    // Prefetch tile t+1 into the other buffer (flies during compute below)
    if (t + 1 < kTiles) {
      const int pbn = pb + 16;
      const unsigned nb = vbase + (unsigned)((cur ^ 1) * BUFF) * 4;
      async_b128(nb + (unsigned)(stc * VST + stp) * 4,
                 g_src + (size_t)stc * kHW + pbn + stp);
      async_b128(nb + (unsigned)((stc + 64) * VST + stp) * 4,
                 g_src + (size_t)(stc + 64) * kHW + pbn + stp);
    }

    const float* vtc = vt + cur * BUFF;

    // S tile: energy[p(16) x q(16)], K dim = ck (16) in 4 chunks of 4
    v8f Sv;
    #pragma unroll
    for (int j = 0; j < 8; ++j) Sv[j] = 0.f;
    #pragma unroll
    for (int c4 = 0; c4 < 4; ++c4) {
      int k0 = c4 * 4 + (hi ? 2 : 0);
      v2f A;  // A[m=p][k]: lane holds M=l16, lane-half selects K {0,1}/{2,3}
      A.x = key2[(size_t)(n * kCK + k0) * kHW + pb + l16];
      A.y = key2[(size_t)(n * kCK + k0 + 1) * kHW + pb + l16];
      Sv = wmma_k4(A, Bq[c4], Sv);
    }

    // Online softmax (per column q = lane%16): reduce over M (regs + swap16)
    float mt = Sv[0];
    #pragma unroll
    for (int j = 1; j < 8; ++j) mt = fmaxf(mt, Sv[j]);
    mt = fmaxf(mt, swap16(mt));
    float mn = fmaxf(m, mt);

    v8f P;
    float st = 0.f;
    #pragma unroll
    for (int j = 0; j < 8; ++j) { P[j] = __expf(Sv[j] - mn); st += P[j]; }
    st += swap16(st);

    float corr = __expf(m - mn);
    l = l * corr + st;
    m = mn;
    #pragma unroll
    for (int tt = 0; tt < 8; ++tt)
      #pragma unroll
      for (int j = 0; j < 8; ++j) O[tt][j] *= corr;

    // Convert P (D layout: M=vgpr+8*hi, N=lane%16) to B-operand chunks over p
    float sxP[8];
    #pragma unroll
    for (int j = 0; j < 8; ++j) sxP[j] = swap16(P[j]);
    v2f Bp[4];
    Bp[0].x = !hi ? P[0]  : sxP[2];  Bp[0].y = !hi ? P[1]  : sxP[3];
    Bp[1].x = !hi ? P[4]  : sxP[6];  Bp[1].y = !hi ? P[5]  : sxP[7];
    Bp[2].x = !hi ? sxP[0] : P[2];   Bp[2].y = !hi ? sxP[1] : P[3];
    Bp[3].x = !hi ? sxP[4] : P[6];   Bp[3].y = !hi ? sxP[5] : P[7];

    // O[c_tile][q] += V(c,p) * P(p,q) : 8 c-tiles x 4 K-chunks
    #pragma unroll
    for (int ct = 0; ct < 8; ++ct) {
      #pragma unroll
      for (int c4 = 0; c4 < 4; ++c4) {
        int p0 = c4 * 4 + (hi ? 2 : 0);
        v2f A;  // A[m=c][k=p]
        A.x = vtc[(ct * 16 + l16) * VST + p0];
        A.y = vtc[(ct * 16 + l16) * VST + p0 + 1];
        O[ct] = wmma_k4(A, Bp[c4], O[ct]);
      }
    }
  }

  // Write unnormalized partial O and per-q (m, l)
  float* Op = Opart + (size_t)(part * kN + n) * kC * kHW;
  #pragma unroll
  for (int ct = 0; ct < 8; ++ct) {
    #pragma unroll
    for (int r = 0; r < 8; ++r) {
      int c = ct * 16 + r + (hi ? 8 : 0);
      Op[(size_t)c * kHW + q] = O[ct][r];
    }
  }
  if (!hi) {
    m_arr[(size_t)(part * kN + n) * kHW + q] = m;
    l_arr[(size_t)(part * kN + n) * kHW + q] = l;
  }
}

// ---------------------------------------------------------------------------
// K5: merge key-partition partials -> out3[n][c][q]
// ---------------------------------------------------------------------------
__global__ __launch_bounds__(256) void combine_kernel(
    const float* __restrict__ Opart, const float* __restrict__ m_arr,
    const float* __restrict__ l_arr, float* __restrict__ out3, int P) {
  int idx = blockIdx.x * 256 + threadIdx.x;  // over N*C*HW
  int q = idx & (kHW - 1);
  int c = (idx >> 12) & (kC - 1);
  int n = idx >> 19;
  float mx = -INFINITY;
  for (int i = 0; i < P; ++i)
    mx = fmaxf(mx, m_arr[(size_t)(i * kN + n) * kHW + q]);
  float denom = 0.f, acc = 0.f;
  for (int i = 0; i < P; ++i) {
    float w = __expf(m_arr[(size_t)(i * kN + n) * kHW + q] - mx);
    denom += w * l_arr[(size_t)(i * kN + n) * kHW + q];
    acc += w * Opart[((size_t)(i * kN + n) * kC + c) * kHW + q];
  }
  out3[idx] = acc / denom;  // idx == (n*C + c)*HW + q
}

// ---------------------------------------------------------------------------
// K6: fuse: out[n][o][q] = ReLU(BN(fuse_w[o][:] . cat(out3, fea)[n][:][q]))
// WMMA GEMM, K=256 in 64 chunks of 4. Wave w -> o tile w; block -> 16 queries.
// ---------------------------------------------------------------------------
__global__ __launch_bounds__(256) void fuse_kernel(
    const float* __restrict__ out3, const float* __restrict__ fea,
    const float* __restrict__ fuse_w, const float* __restrict__ bns,
    const float* __restrict__ bnb, const float* __restrict__ bnm,
    const float* __restrict__ bnv, float* __restrict__ out) {
  const int n   = blockIdx.y;
  const int qb  = blockIdx.x;  // 256 tiles of 16 queries
  const int tid = threadIdx.x;
  const int wave = tid >> 5;
  const int lane = tid & 31;
  const int l16  = lane & 15;
  const bool hi  = lane >= 16;
  const int o_base = wave * 16;
  const int q = qb * 16 + l16;

  v8f acc;
  #pragma unroll
  for (int j = 0; j < 8; ++j) acc[j] = 0.f;

  #pragma unroll 4
  for (int kc = 0; kc < 64; ++kc) {
    int kb = kc * 4 + (hi ? 2 : 0);
    v2f A, B;
    A.x = fuse_w[(size_t)(o_base + l16) * (2 * kC) + kb];
    A.y = fuse_w[(size_t)(o_base + l16) * (2 * kC) + kb + 1];
    const float* src;
    int kk;
    if (kc < 32) { src = out3; kk = kb; }        // first half of cat
    else         { src = fea;  kk = kb - kC; }   // second half of cat
    B.x = src[(size_t)(n * kC + kk) * kHW + q];
    B.y = src[(size_t)(n * kC + kk + 1) * kHW + q];
    acc = wmma_k4(A, B, acc);
  }

  #pragma unroll
  for (int r = 0; r < 8; ++r) {
    int o = o_base + r + (hi ? 8 : 0);
    float inv = bns[o] / sqrtf(bnv[o] + kEPS);
    float v = (acc[r] - bnm[o]) * inv + bnb[o];
    out[(size_t)(n * kC + o) * kHW + q] = fmaxf(v, 0.f);
  }
}

// ---------------------------------------------------------------------------
extern "C" void kernel_launch(void* const* d_in, const int* in_sizes, int n_in,
                              void* d_out, int out_size, void* d_ws,
                              size_t ws_size, hipStream_t stream) {
  const float* query     = (const float*)d_in[0];
  const float* fea       = (const float*)d_in[1];
  const float* key_stack = (const float*)d_in[2];
  const float* fea_stack = (const float*)d_in[3];
  const float* key_w     = (const float*)d_in[4];
  const float* key_b     = (const float*)d_in[5];
  const float* fuse_w    = (const float*)d_in[6];
  const float* bns       = (const float*)d_in[7];
  const float* bnb       = (const float*)d_in[8];
  const float* bnm       = (const float*)d_in[9];
  const float* bnv       = (const float*)d_in[10];
  float* outp = (float*)d_out;

  const size_t att1F = (size_t)kN * kHW * kS;        // 40960
  const size_t out2F = (size_t)kN * kC * kHW;        // 1048576
  const size_t key2F = (size_t)kN * kCK * kHW;       // 131072
  const size_t out3F = out2F;
  const size_t baseF = att1F + out2F + key2F + out3F;

  // Choose key-partition count P (flash-decoding) to fit workspace.
  int P = 8;
  while (P > 1) {
    size_t need = (baseF + (size_t)P * out2F + (size_t)P * kN * kHW * 2) *
                  sizeof(float);
    if (need <= ws_size) break;
    P >>= 1;
  }
  const int kTiles = (kHW / 16) / P;  // key tiles per partition

  float* ws    = (float*)d_ws;
  float* att1  = ws;
  float* out2  = att1 + att1F;
  float* key2  = out2 + out2F;
  float* out3  = key2 + key2F;
  float* Opart = out3 + out3F;
  float* m_arr = Opart + (size_t)P * out2F;
  float* l_arr = m_arr + (size_t)P * kN * kHW;

  att1_kernel<<<(kN * kHW + 255) / 256, 256, 0, stream>>>(query, key_stack,
                                                          att1);
  out2_kernel<<<(kN * kC * kHW) / 256, 256, 0, stream>>>(fea_stack, att1,
                                                         out2);
  key2_kernel<<<(kN * kCK * kHW) / 256, 256, 0, stream>>>(key_w, key_b, out2,
                                                          key2);
  attn2_kernel<<<dim3(kHW / 128, P, kN), 256, 0, stream>>>(
      query, key2, out2, Opart, m_arr, l_arr, kTiles);
  combine_kernel<<<(kN * kC * kHW) / 256, 256, 0, stream>>>(Opart, m_arr,
                                                            l_arr, out3, P);
  fuse_kernel<<<dim3(kHW / 16, kN), 256, 0, stream>>>(out3, fea, fuse_w, bns,
                                                      bnb, bnm, bnv, outp);
}